// XNorCosPosAttention_74758200754303
// MI455X (gfx1250) — compile-verified
//
#include <hip/hip_runtime.h>

// XNorCosPosAttention for gfx1250 (MI455X), fp32 WMMA path.
// Exact linear-attention reformulation:
//   weights = 2*q.k^T + 62  (softmax rows sum to 1)
//   out_i   = (2*q_i.(K^T V) + 62*Vsum) / (2*q_i.Ksum + 62*L)

typedef __attribute__((ext_vector_type(2))) float v2f;
typedef __attribute__((ext_vector_type(4))) float v4f;
typedef __attribute__((ext_vector_type(8))) float v8f;

#define Bx 4
#define Lx 2048
#define Ex 512
#define Hx 8
#define HDx 64
#define Mx (Bx * Lx) /* 8192 */
#define EPSx 1e-6f

#ifndef __has_builtin
#define __has_builtin(x) 0
#endif

#if defined(__HIP_DEVICE_COMPILE__) && \
    __has_builtin(__builtin_amdgcn_global_load_async_to_lds_b128)
#define ASYNC_LDS 1
#else
#define ASYNC_LDS 0
#endif

#if ASYNC_LDS
// The builtin takes pointers to 16-byte int vectors in explicit address
// spaces: global (AS1, printed "__device__") source, LDS (AS3) destination.
typedef int v4i __attribute__((vector_size(16)));
typedef __attribute__((address_space(1))) v4i gas_v4i;
typedef __attribute__((address_space(3))) v4i las_v4i;
#endif

static __device__ __forceinline__ v8f wmma4(v2f a, v2f b, v8f c) {
  // D = A(16x4) * B(4x16) + C(16x16), all f32
  return __builtin_amdgcn_wmma_f32_16x16x4_f32(false, a, false, b, (short)0, c,
                                               false, false);
}

// 16-byte global -> LDS copy; async (ASYNCcnt) when the toolchain exposes it.
static __device__ __forceinline__ void cp_b128(const float* g, float* l) {
#if ASYNC_LDS
  __builtin_amdgcn_global_load_async_to_lds_b128((gas_v4i*)g, (las_v4i*)l, 0,
                                                 0);
#else
  *(v4f*)l = *(const v4f*)g;
#endif
}

static __device__ __forceinline__ void cp_wait() {
#if ASYNC_LDS
#if __has_builtin(__builtin_amdgcn_s_wait_asynccnt)
  __builtin_amdgcn_s_wait_asynccnt(0);
#else
  asm volatile("s_wait_asynccnt 0x0" ::: "memory");
#endif
#endif
}

// ---------------------------------------------------------------------------
// Y(MtotxNtot) = X(MtotxKtot) @ W(NtotxKtot)^T + bias(Ntot)
// Block tile 32(M) x 64(N); K-chunks of 32 double-buffered in LDS with
// 128-bit (async) copies. 8 waves, one 16x16 f32 WMMA accumulator each.
// ---------------------------------------------------------------------------
__global__ __launch_bounds__(256) void gemm_xt_bias(
    const float* __restrict__ X, const float* __restrict__ W,
    const float* __restrict__ bias, float* __restrict__ Y, int Ntot, int Ktot) {
  // stride 36: keeps every float4 slot 16B aligned, 16 distinct banks/row-group
  __shared__ float As[2][32][36];
  __shared__ float Bs[2][64][36];

  const int tid = threadIdx.x;
  const int lane = tid & 31;
  const int w = tid >> 5; // 0..7
  const int tm = w & 1;   // M sub-tile 0..1
  const int tn = w >> 1;  // N sub-tile 0..3
  const int bm = blockIdx.x * 32;
  const int bn = blockIdx.y * 64;

  const int frow = tid >> 3;      // fill row 0..31
  const int fc4 = (tid & 7) * 4;  // fill col (float4 granularity)

  v8f acc = {};
  const int ar = tm * 16 + (lane & 15);
  const int br = tn * 16 + (lane & 15);
  const int klo = 2 * (lane >> 4); // K lane offset per ISA A/B layout

  // issue staging copies for K-chunk k0 into buffer buf (3 x b128 per thread)
  auto issue = [&](int k0, int buf) {
    cp_b128(&X[(size_t)(bm + frow) * Ktot + k0 + fc4], &As[buf][frow][fc4]);
    cp_b128(&W[(size_t)(bn + frow) * Ktot + k0 + fc4], &Bs[buf][frow][fc4]);
    cp_b128(&W[(size_t)(bn + 32 + frow) * Ktot + k0 + fc4],
            &Bs[buf][32 + frow][fc4]);
  };

  const int nk = Ktot >> 5;
  issue(0, 0);
  for (int i = 0; i < nk; ++i) {
    cp_wait();
    __syncthreads(); // chunk i resident for all waves
    if (i + 1 < nk) issue((i + 1) * 32, (i + 1) & 1); // overlap next copy
    const float* Ab = &As[i & 1][ar][klo];
    const float* Bb = &Bs[i & 1][br][klo];
#pragma unroll
    for (int ks = 0; ks < 32; ks += 4) {
      v2f a = *(const v2f*)(Ab + ks);
      v2f b = *(const v2f*)(Bb + ks);
      acc = wmma4(a, b, acc);
    }
    __syncthreads(); // all waves done reading buffer i&1
  }

  const int col = bn + tn * 16 + (lane & 15);
  const float bv = bias[col];
  const int rbase = bm + tm * 16 + 8 * (lane >> 4);
#pragma unroll
  for (int v = 0; v < 8; ++v)
    Y[(size_t)(rbase + v) * Ntot + col] = acc[v] + bv;
}

// ---------------------------------------------------------------------------
// In-place softmax over each 64-wide head block of P0 (Q) and P1 (K).
// One wave per (row, head) group; lane holds 2 contiguous floats.
// ---------------------------------------------------------------------------
__global__ __launch_bounds__(256) void head_softmax(float* __restrict__ P0,
                                                    float* __restrict__ P1) {
  const int lane = threadIdx.x & 31;
  const int gid = blockIdx.x * 8 + (threadIdx.x >> 5); // 0..131071
  float* P = (gid < Mx * Hx) ? P0 : P1;
  const int rem = gid & (Mx * Hx - 1);
  const int m = rem >> 3, h = rem & 7;
  float* p = P + (size_t)m * Ex + h * HDx + 2 * lane;

  v2f x = *(const v2f*)p;
  float mx = fmaxf(x.x, x.y);
#pragma unroll
  for (int off = 16; off; off >>= 1) mx = fmaxf(mx, __shfl_xor(mx, off, 32));
  float e0 = expf(x.x - mx), e1 = expf(x.y - mx);
  float s = e0 + e1;
#pragma unroll
  for (int off = 16; off; off >>= 1) s += __shfl_xor(s, off, 32);
  const float inv = 1.0f / s;
  v2f r;
  r.x = e0 * inv;
  r.y = e1 * inv;
  *(v2f*)p = r;
}

// ---------------------------------------------------------------------------
// KV[n] = K_n^T @ V_n (64x64) per batch-head n. One block per n, 16 waves
// covering the 4x4 grid of 16x16 tiles; K-dim = L = 2048 in steps of 4.
// ---------------------------------------------------------------------------
__global__ __launch_bounds__(512) void kv_outer(const float* __restrict__ Km,
                                                const float* __restrict__ Vm,
                                                float* __restrict__ KV) {
  const int n = blockIdx.x; // 0..31  (n = b*H + h)
  const int b = n >> 3, hoff = (n & 7) * HDx;
  const int lane = threadIdx.x & 31;
  const int w = threadIdx.x >> 5; // 0..15
  const int td1 = w & 3, td2 = w >> 2;
  const int d1 = td1 * 16 + (lane & 15);
  const int d2 = td2 * 16 + (lane & 15);
  const int klo = 2 * (lane >> 4);

  v8f acc = {};
  for (int l0 = 0; l0 < Lx; l0 += 4) {
    const int kb = l0 + klo;
    const size_t r0 = ((size_t)kb * Bx + b) * Ex + hoff;
    const size_t r1 = ((size_t)(kb + 1) * Bx + b) * Ex + hoff;
    v2f a, bb;
    a.x = Km[r0 + d1];
    a.y = Km[r1 + d1];
    bb.x = Vm[r0 + d2];
    bb.y = Vm[r1 + d2];
    acc = wmma4(a, bb, acc);
  }
#pragma unroll
  for (int v = 0; v < 8; ++v) {
    const int dd1 = td1 * 16 + v + 8 * (lane >> 4);
    KV[((size_t)n * 64 + dd1) * 64 + d2] = acc[v];
  }
}

// Ksum[n][d] = sum_l K_n[l][d];  Vsum[n][d] = sum_l V_n[l][d]
__global__ __launch_bounds__(128) void col_sums(const float* __restrict__ Km,
                                                const float* __restrict__ Vm,
                                                float* __restrict__ Ksum,
                                                float* __restrict__ Vsum) {
  const int n = blockIdx.x;
  const int b = n >> 3, hoff = (n & 7) * HDx;
  const int tid = threadIdx.x;
  const float* src = (tid < 64) ? Km : Vm;
  const int d = tid & 63;
  float s = 0.f;
  for (int l = 0; l < Lx; ++l) s += src[((size_t)l * Bx + b) * Ex + hoff + d];
  if (tid < 64)
    Ksum[n * 64 + d] = s;
  else
    Vsum[n * 64 + d] = s;
}

// den[m*8+h] = max(2*q_row.Ksum + 62*L, EPS); one wave per (m,h)
__global__ __launch_bounds__(256) void den_kernel(const float* __restrict__ Q,
                                                  const float* __restrict__ Ksum,
                                                  float* __restrict__ den) {
  const int lane = threadIdx.x & 31;
  const int gid = blockIdx.x * 8 + (threadIdx.x >> 5); // m*8 + h
  const int m = gid >> 3, h = gid & 7;
  const int n = (m & 3) * Hx + h; // b = m % B
  v2f q = *(const v2f*)&Q[(size_t)m * Ex + h * HDx + 2 * lane];
  v2f ks = *(const v2f*)&Ksum[n * 64 + 2 * lane];
  float p = q.x * ks.x + q.y * ks.y;
#pragma unroll
  for (int off = 16; off; off >>= 1) p += __shfl_xor(p, off, 32);
  if (lane == 0) den[gid] = fmaxf(2.f * p + 62.f * (float)Lx, EPSx);
}

// ---------------------------------------------------------------------------
// A_n = (2 * Q_n @ KV_n + 62*Vsum) / den   -> scattered back into 8192x512 A
// grid: (128 row-tiles, 32 heads); 4 waves = 4 column tiles of 16.
// ---------------------------------------------------------------------------
__global__ __launch_bounds__(128) void attn_apply(
    const float* __restrict__ Q, const float* __restrict__ KV,
    const float* __restrict__ Vsum, const float* __restrict__ den,
    float* __restrict__ A) {
  const int n = blockIdx.y;
  const int b = n >> 3, h = n & 7, hoff = h * HDx;
  const int it = blockIdx.x; // 16-row tile index within L
  const int lane = threadIdx.x & 31;
  const int w = threadIdx.x >> 5; // column tile 0..3
  const int klo = 2 * (lane >> 4);
  const int d2 = w * 16 + (lane & 15);

  const int arow = it * 16 + (lane & 15); // sequence position for A fragment
  const size_t qrow = ((size_t)arow * Bx + b) * Ex + hoff;

  v8f acc = {};
#pragma unroll
  for (int k0 = 0; k0 < HDx; k0 += 4) {
    const int kb = k0 + klo;
    v2f a = *(const v2f*)&Q[qrow + kb];
    v2f bb;
    bb.x = KV[((size_t)n * 64 + kb) * 64 + d2];
    bb.y = KV[((size_t)n * 64 + kb + 1) * 64 + d2];
    acc = wmma4(a, bb, acc);
  }

  const float vs = Vsum[n * 64 + d2];
#pragma unroll
  for (int v = 0; v < 8; ++v) {
    const int i = it * 16 + v + 8 * (lane >> 4);
    const int m = i * Bx + b;
    const float dn = den[m * 8 + h];
    A[(size_t)m * Ex + hoff + d2] = (2.f * acc[v] + 62.f * vs) / dn;
  }
}

// ---------------------------------------------------------------------------
extern "C" void kernel_launch(void* const* d_in, const int* in_sizes, int n_in,
                              void* d_out, int out_size, void* d_ws,
                              size_t ws_size, hipStream_t stream) {
  (void)in_sizes; (void)n_in; (void)out_size; (void)ws_size;
  const float* query = (const float*)d_in[0];
  const float* Wq = (const float*)d_in[1];
  const float* bq = (const float*)d_in[2];
  const float* Wk = (const float*)d_in[3];
  const float* bk = (const float*)d_in[4];
  const float* Wv = (const float*)d_in[5];
  const float* bv = (const float*)d_in[6];
  const float* Wo = (const float*)d_in[7];
  const float* bo = (const float*)d_in[8];

  float* ws = (float*)d_ws;
  const size_t MAT = (size_t)Mx * Ex; // 4,194,304 floats
  float* Q    = ws;
  float* Kmat = ws + MAT;
  float* Vmat = ws + 2 * MAT;
  float* Amat = ws + 3 * MAT;
  float* KV   = ws + 4 * MAT;          // 32*64*64
  float* Ksum = KV + 32 * 64 * 64;     // 32*64
  float* Vsum = Ksum + 32 * 64;        // 32*64
  float* den  = Vsum + 32 * 64;        // 8192*8

  dim3 gg(Mx / 32, Ex / 64), gb(256);
  gemm_xt_bias<<<gg, gb, 0, stream>>>(query, Wq, bq, Q, Ex, Ex);
  gemm_xt_bias<<<gg, gb, 0, stream>>>(query, Wk, bk, Kmat, Ex, Ex);
  gemm_xt_bias<<<gg, gb, 0, stream>>>(query, Wv, bv, Vmat, Ex, Ex);

  head_softmax<<<(2 * Mx * Hx) / 8, 256, 0, stream>>>(Q, Kmat);

  kv_outer<<<32, 512, 0, stream>>>(Kmat, Vmat, KV);
  col_sums<<<32, 128, 0, stream>>>(Kmat, Vmat, Ksum, Vsum);
  den_kernel<<<(Mx * Hx) / 8, 256, 0, stream>>>(Q, Ksum, den);
  attn_apply<<<dim3(Lx / 16, Bx * Hx), 128, 0, stream>>>(Q, KV, Vsum, den, Amat);

  gemm_xt_bias<<<gg, gb, 0, stream>>>(Amat, Wo, bo, (float*)d_out, Ex, Ex);
}